// GAT_1649267441817
// MI455X (gfx1250) — compile-verified
//
#include <hip/hip_runtime.h>

// ---------------- problem constants (match reference) ----------------
constexpr int   N_    = 50000;
constexpr int   E_    = 600000;
constexpr int   EN_   = E_ + N_;     // edges + self loops
constexpr int   D_    = 128;         // D_IN == D_H
constexpr int   G_    = 64;
constexpr int   DOUT_ = 64;
constexpr int   L_    = 3;
#define SLOPE 0.2f

typedef __attribute__((ext_vector_type(2))) float v2f;
typedef __attribute__((ext_vector_type(8))) float v8f;

// ---- monotonic float<->uint key mapping for atomicMax over floats ----
__device__ __forceinline__ unsigned fkey(float f) {
  unsigned b = __float_as_uint(f);
  return (b & 0x80000000u) ? ~b : (b | 0x80000000u);
}
__device__ __forceinline__ float keyf(unsigned k) {
  return __uint_as_float((k & 0x80000000u) ? (k ^ 0x80000000u) : ~k);
}

// ---------------------------------------------------------------------
// Full-precision WMMA GEMM: C[M x Nc] = A[M x K] * W[K x Nc] (+ bias)
// One wave per 16x16 output tile, K stepped by 4 with v_wmma_f32_16x16x4_f32.
// M, Nc multiples of 16; K multiple of 4.
// ---------------------------------------------------------------------
__global__ void gemm_f32_wmma(const float* __restrict__ A,
                              const float* __restrict__ W,
                              const float* __restrict__ bias,
                              float* __restrict__ C,
                              int M, int K, int Nc) {
  const int wave  = (blockIdx.x * blockDim.x + threadIdx.x) >> 5;
  const int lane  = threadIdx.x & 31;
  const int tilesN = Nc >> 4;
  const int tiles  = (M >> 4) * tilesN;
  if (wave >= tiles) return;                 // whole wave exits together
  const int tileM = wave / tilesN;
  const int tileN = wave % tilesN;

  const int mn   = lane & 15;                // A-row index == B/C/D column index
  const int half = lane >> 4;                // K sub-slot selector

  const float* Arow = A + (size_t)(tileM * 16 + mn) * K;
  const float* Wcol = W + (size_t)(tileN * 16 + mn);

  v8f acc = {};
  for (int k = 0; k < K; k += 4) {
    // A 16x4 tile: element j = A[m][k + 2*half + j]  (8B aligned)
    v2f a = *(const v2f*)(Arow + k + 2 * half);
    // B 4x16 tile: element j = W[k + 2*half + j][n]
    v2f b;
    b.x = Wcol[(size_t)(k + 2 * half) * Nc];
    b.y = Wcol[(size_t)(k + 2 * half + 1) * Nc];
    acc = __builtin_amdgcn_wmma_f32_16x16x4_f32(false, a, false, b,
                                                (short)0, acc, false, false);
  }

  const float bc = bias ? bias[tileN * 16 + mn] : 0.0f;
  float* Crow = C + (size_t)(tileM * 16) * Nc + tileN * 16 + mn;
#pragma unroll
  for (int v = 0; v < 8; ++v) {
    int m = v + 8 * half;                    // D VGPR v holds rows v / v+8
    Crow[(size_t)m * Nc] = acc[v] + bc;
  }
}

// ---------------------------------------------------------------------
// Per-node attention scores: es[n] = hp[n,:].a_src ; ed[n] = hp[n,:].a_dst
// One wave per node, shuffle reduction.
// ---------------------------------------------------------------------
__global__ void node_scores(const float* __restrict__ hp,
                            const float* __restrict__ asrc,
                            const float* __restrict__ adst,
                            float* __restrict__ es, float* __restrict__ ed) {
  const int wave = (blockIdx.x * blockDim.x + threadIdx.x) >> 5;
  const int lane = threadIdx.x & 31;
  if (wave >= N_) return;
  const float* row = hp + (size_t)wave * D_;
  float s0 = 0.f, s1 = 0.f;
#pragma unroll
  for (int c = lane; c < D_; c += 32) {
    float v = row[c];
    s0 += v * asrc[c];
    s1 += v * adst[c];
  }
#pragma unroll
  for (int o = 16; o > 0; o >>= 1) {
    s0 += __shfl_down(s0, o);
    s1 += __shfl_down(s1, o);
  }
  if (lane == 0) { es[wave] = s0; ed[wave] = s1; }
}

// Per-layer reset: acc = 0, segment max key = 0 (-inf-ish), z = 0.
__global__ void layer_init(float* __restrict__ acc, unsigned* __restrict__ mkey,
                           float* __restrict__ z) {
  const int idx = blockIdx.x * blockDim.x + threadIdx.x;
  if (idx < N_ * D_) acc[idx] = 0.f;
  if (idx < N_) { mkey[idx] = 0u; z[idx] = 0.f; }
}

// Edge pass 1: e = leaky_relu(es[src]+ed[dst]); segment max via key atomicMax.
__global__ void edge_max(const int* __restrict__ src, const int* __restrict__ dst,
                         const float* __restrict__ es, const float* __restrict__ ed,
                         float* __restrict__ ebuf, unsigned* __restrict__ mkey) {
  const int e = blockIdx.x * blockDim.x + threadIdx.x;
  if (e >= EN_) return;
  const int s = (e < E_) ? src[e] : (e - E_);   // self-loops appended
  const int d = (e < E_) ? dst[e] : (e - E_);
  float v = es[s] + ed[d];
  v = (v > 0.f) ? v : SLOPE * v;
  ebuf[e] = v;
  atomicMax(&mkey[d], fkey(v));
}

// Edge pass 2: ex = exp(e - m[dst]); z[dst] += ex.
__global__ void edge_exp(const int* __restrict__ dst,
                         const float* __restrict__ ebuf,
                         const unsigned* __restrict__ mkey,
                         float* __restrict__ exbuf, float* __restrict__ z) {
  const int e = blockIdx.x * blockDim.x + threadIdx.x;
  if (e >= EN_) return;
  const int d = (e < E_) ? dst[e] : (e - E_);
  const float x = expf(ebuf[e] - keyf(mkey[d]));
  exbuf[e] = x;
  atomicAdd(&z[d], x);
}

// Edge pass 3: acc[dst,:] += (ex/z[dst]) * hp[src,:]. One wave per edge.
__global__ void edge_scatter(const int* __restrict__ src, const int* __restrict__ dst,
                             const float* __restrict__ hp,
                             const float* __restrict__ exbuf,
                             const float* __restrict__ z,
                             float* __restrict__ acc) {
  const int wave = (blockIdx.x * blockDim.x + threadIdx.x) >> 5;
  const int lane = threadIdx.x & 31;
  if (wave >= EN_) return;
  const int e = wave;
  const int s = (e < E_) ? src[e] : (e - E_);
  const int d = (e < E_) ? dst[e] : (e - E_);
  const float alpha = exbuf[e] / z[d];
  const float4 hv = *(const float4*)(hp + (size_t)s * D_ + lane * 4);
  float* ap = acc + (size_t)d * D_ + lane * 4;
  atomicAdd(ap + 0, alpha * hv.x);
  atomicAdd(ap + 1, alpha * hv.y);
  atomicAdd(ap + 2, alpha * hv.z);
  atomicAdd(ap + 3, alpha * hv.w);
}

// h = relu(acc + bg) + h   (residual, in place)
__global__ void post_residual(float* __restrict__ h, const float* __restrict__ acc,
                              const float* __restrict__ bgl) {
  const int idx = blockIdx.x * blockDim.x + threadIdx.x;
  if (idx >= N_ * D_) return;
  float v = acc[idx] + bgl[idx & (D_ - 1)];
  h[idx] += (v > 0.f) ? v : 0.f;
}

__global__ void zero_pooled(float* __restrict__ pooled) {
  const int idx = blockIdx.x * blockDim.x + threadIdx.x;
  if (idx < G_ * D_) pooled[idx] = 0.f;
}

// Sorted-batch global_add_pool: per-column running sum, flush on graph change.
__global__ void pool_sum(const float* __restrict__ h, const int* __restrict__ batch,
                         float* __restrict__ pooled) {
  const int c = threadIdx.x;                 // 128 threads per block
  const int start = blockIdx.x * 128;
  if (start >= N_) return;
  const int end = (start + 128 < N_) ? start + 128 : N_;
  int cur = batch[start];
  float sum = 0.f;
  for (int n = start; n < end; ++n) {
    int b = batch[n];
    if (b != cur) {
      atomicAdd(&pooled[(size_t)cur * D_ + c], sum);
      sum = 0.f; cur = b;
    }
    sum += h[(size_t)n * D_ + c];
  }
  atomicAdd(&pooled[(size_t)cur * D_ + c], sum);
}

// ---------------------------------------------------------------------
extern "C" void kernel_launch(void* const* d_in, const int* in_sizes, int n_in,
                              void* d_out, int out_size, void* d_ws, size_t ws_size,
                              hipStream_t stream) {
  const float* x     = (const float*)d_in[0];
  const int*   ei    = (const int*)  d_in[1];   // [2, E] flat
  const int*   batch = (const int*)  d_in[2];
  const float* W1    = (const float*)d_in[3];
  const float* b1    = (const float*)d_in[4];
  const float* Wg    = (const float*)d_in[5];   // [L,128,128]
  const float* a_src = (const float*)d_in[6];   // [L,128]
  const float* a_dst = (const float*)d_in[7];
  const float* bg    = (const float*)d_in[8];
  const float* W2    = (const float*)d_in[9];
  const float* b2    = (const float*)d_in[10];
  float* out = (float*)d_out;

  const int* src = ei;
  const int* dst = ei + E_;

  // ---- workspace carve-up (256B aligned) ----
  char* ws = (char*)d_ws;
  auto align256 = [](size_t v) { return (v + 255) & ~(size_t)255; };
  size_t off = 0;
  const size_t NB = (size_t)N_ * D_ * sizeof(float);
  float*    h      = (float*)(ws + off); off = align256(off + NB);
  float*    hp     = (float*)(ws + off); off = align256(off + NB);
  float*    acc    = (float*)(ws + off); off = align256(off + NB);
  float*    es     = (float*)(ws + off); off = align256(off + (size_t)N_ * 4);
  float*    ed     = (float*)(ws + off); off = align256(off + (size_t)N_ * 4);
  float*    ebuf   = (float*)(ws + off); off = align256(off + (size_t)EN_ * 4);
  float*    exbuf  = (float*)(ws + off); off = align256(off + (size_t)EN_ * 4);
  unsigned* mkey   = (unsigned*)(ws + off); off = align256(off + (size_t)N_ * 4);
  float*    z      = (float*)(ws + off); off = align256(off + (size_t)N_ * 4);
  float*    pooled = (float*)(ws + off); off = align256(off + (size_t)G_ * D_ * 4);

  const int BLK = 256;
  const int gemmBlocksBig = ((N_ / 16) * (D_ / 16) * 32 + BLK - 1) / BLK;   // 3125
  const int ndBlocks      = (N_ * D_ + BLK - 1) / BLK;                      // 25000
  const int nodeBlocks    = (N_ * 32 + BLK - 1) / BLK;                      // 6250
  const int edgeBlocks    = (EN_ + BLK - 1) / BLK;                          // 2540
  const int scatBlocks    = ((size_t)EN_ * 32 + BLK - 1) / BLK;             // 81250
  const int poolBlocks    = (N_ + 127) / 128;                               // 391

  // mlp1: h = x @ W1 + b1
  gemm_f32_wmma<<<gemmBlocksBig, BLK, 0, stream>>>(x, W1, b1, h, N_, D_, D_);

  for (int l = 0; l < L_; ++l) {
    const float* Wl  = Wg    + (size_t)l * D_ * D_;
    const float* asl = a_src + (size_t)l * D_;
    const float* adl = a_dst + (size_t)l * D_;
    const float* bgl = bg    + (size_t)l * D_;

    layer_init  <<<ndBlocks,   BLK, 0, stream>>>(acc, mkey, z);
    gemm_f32_wmma<<<gemmBlocksBig, BLK, 0, stream>>>(h, Wl, nullptr, hp, N_, D_, D_);
    node_scores <<<nodeBlocks, BLK, 0, stream>>>(hp, asl, adl, es, ed);
    edge_max    <<<edgeBlocks, BLK, 0, stream>>>(src, dst, es, ed, ebuf, mkey);
    edge_exp    <<<edgeBlocks, BLK, 0, stream>>>(dst, ebuf, mkey, exbuf, z);
    edge_scatter<<<scatBlocks, BLK, 0, stream>>>(src, dst, hp, exbuf, z, acc);
    post_residual<<<ndBlocks,  BLK, 0, stream>>>(h, acc, bgl);
  }

  zero_pooled<<<(G_ * D_ + BLK - 1) / BLK, BLK, 0, stream>>>(pooled);
  pool_sum   <<<poolBlocks, 128, 0, stream>>>(h, batch, pooled);

  // out = pooled @ W2 + b2   (64x128 @ 128x64, 16 WMMA tiles)
  gemm_f32_wmma<<<2, BLK, 0, stream>>>(pooled, W2, b2, out, G_, D_, DOUT_);
}